// VTDM_11287174053960
// MI455X (gfx1250) — compile-verified
//
#include <hip/hip_runtime.h>
#include <stdint.h>

// VTDM two-compartment PK scan, MI455X (gfx1250, wave32).
// One wave per batch row; lanes = 32 consecutive timesteps per tile.
// Affine maps (alpha, beta) for both states are scanned across the wave
// (Kogge-Stone, 5 shuffle levels), carry broadcast from lane 31.
// Inputs streamed via double-buffered global_load_async_to_lds_b32 (ASYNCcnt).

#define T_LEN            8192
#define TILE             32
#define WAVES_PER_BLOCK  8
#define BLOCK_THREADS    (WAVES_PER_BLOCK * 32)

__device__ __forceinline__ void async_load_f32(const float* __restrict__ base,
                                               uint32_t elem_off,
                                               uint32_t lds_byte_addr) {
  // GVS mode: mem_addr = SGPR_base + VGPR_byte_offset; per-lane LDS dest addr.
  asm volatile("global_load_async_to_lds_b32 %0, %1, %2"
               :
               : "v"(lds_byte_addr), "v"(elem_off << 2), "s"(base)
               : "memory");
}

__global__ __launch_bounds__(BLOCK_THREADS)
void vtdm_scan_kernel(const float* __restrict__ dose,
                      const float* __restrict__ tdiff,
                      const float* __restrict__ ccl,
                      const float* __restrict__ eta1,
                      const float* __restrict__ eta2,
                      const float* __restrict__ eta3,
                      float* __restrict__ out, int B) {
  // [wave][double-buffer][array: dose/tdiff/ccl][lane]
  __shared__ float lds[WAVES_PER_BLOCK][2][3][TILE];

  const int wave = threadIdx.x >> 5;
  const int lane = threadIdx.x & 31;
  const int row  = blockIdx.x * WAVES_PER_BLOCK + wave;
  if (row >= B) return;

  // ---- scalar PK constants (uniform: broadcast scalar loads) ----
  const float e1 = eta1[0], e2 = eta2[0], e3 = eta3[0];
  const float V1      = 33.1f * __expf(e1);
  const float Q       = 6.99f * __expf(e3);
  const float k12     = Q / V1;
  const float k21     = Q / 48.3f;
  const float Rk12    = (1000.0f / V1) * k12;
  const float inv_k12 = 1.0f / k12;
  const float kf      = 0.0396f * __expf(e2) / V1;   // k10 = ccl * kf

  const uint32_t rowbase = (uint32_t)row * (uint32_t)T_LEN;  // < 2^25, fits u32

  auto issue_tile = [&](int tileIdx, int buf) {
    const uint32_t off = rowbase + (uint32_t)(tileIdx * TILE + lane);
    async_load_f32(dose,  off, (uint32_t)(uintptr_t)&lds[wave][buf][0][lane]);
    async_load_f32(tdiff, off, (uint32_t)(uintptr_t)&lds[wave][buf][1][lane]);
    async_load_f32(ccl,   off, (uint32_t)(uintptr_t)&lds[wave][buf][2][lane]);
  };

  const int ntiles = T_LEN / TILE;   // 256
  issue_tile(0, 0);                  // prime the pipeline (ASYNCcnt = 3)

  float Ccar = 0.0f, Dcar = 0.0f;

  for (int tile = 0; tile < ntiles; ++tile) {
    // Prefetch next tile into the other buffer (last iter re-fetches harmlessly).
    const int nxt = (tile + 1 < ntiles) ? (tile + 1) : tile;
    issue_tile(nxt, (tile + 1) & 1);

    // Wait until the CURRENT tile's 3 loads landed in LDS; 3 newer stay in flight.
    asm volatile("s_wait_asynccnt 0x3" ::: "memory");

    const int buf  = tile & 1;
    const float d  = lds[wave][buf][0][lane];
    const float td = lds[wave][buf][1][lane];
    const float cc = lds[wave][buf][2][lane];

    // ---- element-wise eigen-structure (all lane-parallel) ----
    const float k10   = cc * kf;
    const float s     = k10 + k12 + k21;
    const float delta = sqrtf(fmaf(s, s, -4.0f * k10 * k21));
    const float l1    = -0.5f * (s + delta);
    const float l2    =  0.5f * (delta - s);
    const float c1v   = __fdividef(-Rk12, l1 * delta);
    const float c2v   = __fdividef( Rk12, l2 * delta);
    const float c3v   = (l1 + k21) * inv_k12;
    const float c4v   = (l2 + k21) * inv_k12;

    // Affine map per timestep: state' = a*state + b
    float aC = __expf(l1 * td);
    float aD = __expf(l2 * td);
    const float E1b = __expf(l1 * (td - d));
    const float E2b = __expf(l2 * (td - d));
    const bool dosed = (d != 0.0f);
    float bC = dosed ? c1v * (aC - E1b) : 0.0f;
    float bD = dosed ? c2v * (aD - E2b) : 0.0f;

    // ---- wave32 inclusive affine scan (Kogge-Stone, 5 levels) ----
    #pragma unroll
    for (int dlt = 1; dlt < 32; dlt <<= 1) {
      const float aCp = __shfl_up(aC, dlt, 32);
      const float bCp = __shfl_up(bC, dlt, 32);
      const float aDp = __shfl_up(aD, dlt, 32);
      const float bDp = __shfl_up(bD, dlt, 32);
      if (lane >= dlt) {            // compose: (a,b) = (a,b) ∘ (ap,bp)
        bC = fmaf(aC, bCp, bC);
        aC *= aCp;
        bD = fmaf(aD, bDp, bD);
        aD *= aDp;
      }
    }

    // Apply scanned map to the incoming carry; emit output.
    const float Ct = fmaf(aC, Ccar, bC);
    const float Dt = fmaf(aD, Dcar, bD);
    const float A  = Ct * c3v + Dt * c4v;
    __builtin_nontemporal_store(A, &out[rowbase + (uint32_t)(tile * TILE + lane)]);

    // New carry = state after the last timestep of this tile.
    Ccar = __shfl(Ct, 31, 32);
    Dcar = __shfl(Dt, 31, 32);
  }
}

extern "C" void kernel_launch(void* const* d_in, const int* in_sizes, int n_in,
                              void* d_out, int out_size, void* d_ws, size_t ws_size,
                              hipStream_t stream) {
  (void)n_in; (void)out_size; (void)d_ws; (void)ws_size;
  const float* dose  = (const float*)d_in[0];
  const float* tdiff = (const float*)d_in[1];
  const float* ccl   = (const float*)d_in[2];
  const float* eta1  = (const float*)d_in[3];
  const float* eta2  = (const float*)d_in[4];
  const float* eta3  = (const float*)d_in[5];
  float* out = (float*)d_out;

  const int B = in_sizes[0] / T_LEN;                       // 4096
  const int blocks = (B + WAVES_PER_BLOCK - 1) / WAVES_PER_BLOCK;
  vtdm_scan_kernel<<<blocks, BLOCK_THREADS, 0, stream>>>(
      dose, tdiff, ccl, eta1, eta2, eta3, out, B);
}